// NetCrossing_53455162966425
// MI455X (gfx1250) — compile-verified
//
#include <hip/hip_runtime.h>
#include <hip/hip_bf16.h>

#ifndef __has_builtin
#define __has_builtin(x) 0
#endif

#define NC_THREADS 256
#define NC_SIG_K   5.0f   /* LAMBDA / SIGMA = 10 / 2 */
#define NC_MU      1.0f

typedef __attribute__((ext_vector_type(2))) float v2f;
typedef __attribute__((ext_vector_type(8))) float v8f;

#if __has_builtin(__builtin_amdgcn_global_load_async_to_lds_b32)
#define NC_ASYNC 1
#else
#define NC_ASYNC 0
#endif

typedef __attribute__((address_space(1))) int g_int;
typedef __attribute__((address_space(3))) int l_int;

__device__ __forceinline__ float nc_sig(float x) {
    // sigmoid(5x), stable for large |x|: exp(-5x)->inf => 0, ->0 => 1
    return 1.0f / (1.0f + __expf(-NC_SIG_K * x));
}

__device__ __forceinline__ float nc_opp(float u, float v) {
    // sig(u)*sig(-v) + sig(-u)*sig(v) == su + sv - 2*su*sv  (sig(-x)=1-sig(x))
    float su = nc_sig(u), sv = nc_sig(v);
    return su + sv - 2.0f * su * sv;
}

__global__ __launch_bounds__(NC_THREADS)
void nc_main_kernel(const float* __restrict__ pos,
                    const int* __restrict__ flat_netpin,
                    const int* __restrict__ netpin_start,
                    const unsigned char* __restrict__ net_mask, // jax bool -> 1 byte
                    const int* __restrict__ max_degree_p,
                    int num_nets, int num_pins,
                    float* __restrict__ block_partials) {
    __shared__ int   s_start[NC_THREADS + 1];
    __shared__ float s_wsum[NC_THREADS / 32];

    const int t  = threadIdx.x;
    const int n0 = blockIdx.x * NC_THREADS;

    // ---- stage netpin_start[n0 .. n0+256] into LDS (async copy path) ----
#if NC_ASYNC
    {
        int gi = n0 + t; if (gi > num_nets) gi = num_nets;
        __builtin_amdgcn_global_load_async_to_lds_b32(
            (g_int*)(netpin_start + gi), (l_int*)&s_start[t], 0, 0);
        if (t == NC_THREADS - 1) {
            int gj = n0 + NC_THREADS; if (gj > num_nets) gj = num_nets;
            __builtin_amdgcn_global_load_async_to_lds_b32(
                (g_int*)(netpin_start + gj), (l_int*)&s_start[NC_THREADS], 0, 0);
        }
#if __has_builtin(__builtin_amdgcn_s_wait_asynccnt)
        __builtin_amdgcn_s_wait_asynccnt(0);
#else
        asm volatile("s_wait_asynccnt 0" ::: "memory");
#endif
    }
#else
    {
        int gi = n0 + t; if (gi > num_nets) gi = num_nets;
        s_start[t] = netpin_start[gi];
        if (t == NC_THREADS - 1) {
            int gj = n0 + NC_THREADS; if (gj > num_nets) gj = num_nets;
            s_start[NC_THREADS] = netpin_start[gj];
        }
    }
#endif
    __syncthreads();

    // ---- one thread per net ----
    float partial = 0.0f;
    const int n = n0 + t;
    if (n < num_nets && net_mask[n] != 0) {
        const int start = s_start[t];
        const int end   = s_start[t + 1];
        int D = max_degree_p[0]; if (D > 8) D = 8;
        int d = end - start;     if (d > D) d = D;
        if (d >= 4) {  // need >= 3 segments for a non-adjacent pair
            float px[8], py[8];
#pragma unroll
            for (int k = 0; k < 8; ++k) {
                if (k < d) {
                    int pid = flat_netpin[start + k];
                    px[k] = pos[pid];
                    py[k] = pos[num_pins + pid];
                } else { px[k] = 0.0f; py[k] = 0.0f; }
            }
            const int S = d - 1;           // segments in this net
#pragma unroll
            for (int i = 0; i < 5; ++i) {  // i <= 4 since j <= 6
#pragma unroll
                for (int j = i + 2; j < 7; ++j) {
                    if (j < S) {
                        float ax = px[i],     ay = py[i];
                        float bx = px[i + 1], by = py[i + 1];
                        float cx = px[j],     cy = py[j];
                        float ex = px[j + 1], ey = py[j + 1];
                        float d1 = (cx - ax) * (ey - ay) - (cy - ay) * (ex - ax);
                        float d2 = (cx - bx) * (ey - by) - (cy - by) * (ex - bx);
                        float d3 = (bx - ax) * (cy - ay) - (by - ay) * (cx - ax);
                        float d4 = (bx - ax) * (ey - ay) - (by - ay) * (ex - ax);
                        partial += nc_opp(d1, d2) * nc_opp(d3, d4);
                    }
                }
            }
        }
    }

    // ---- deterministic block reduction (wave32 shuffles -> LDS) ----
    float v = partial;
#pragma unroll
    for (int off = 16; off > 0; off >>= 1) v += __shfl_down(v, off, 32);
    const int lane = t & 31, wid = t >> 5;
    if (lane == 0) s_wsum[wid] = v;
    __syncthreads();
    if (t == 0) {
        float s = 0.0f;
#pragma unroll
        for (int w = 0; w < NC_THREADS / 32; ++w) s += s_wsum[w];
        block_partials[blockIdx.x] = s;
    }
}

// Final 256->1 reduction. The 256 per-thread partials are summed with four
// chained V_WMMA_F32_16X16X4_F32 (B = ones): D[m][n] = sum_k A[m][k] + C[m][n].
// Each lane feeds 2 A-values per chunk (4 chunks = 8 values/lane = 256 values);
// column N=0 of D lives in lanes 0 (rows 0..7) and 16 (rows 8..15).
__global__ __launch_bounds__(NC_THREADS)
void nc_final_kernel(const float* __restrict__ partials, int n,
                     float* __restrict__ out, float mu) {
    __shared__ float s_val[NC_THREADS];
    const int t = threadIdx.x;
    float acc = 0.0f;
    for (int i = t; i < n; i += NC_THREADS) acc += partials[i];  // fixed order
    s_val[t] = acc;
    __syncthreads();
    if (t < 32) {
#if __has_builtin(__builtin_amdgcn_wmma_f32_16x16x4_f32)
        v8f c = {};
        const v2f ones = {1.0f, 1.0f};
#pragma unroll
        for (int k = 0; k < 4; ++k) {
            v2f a;
            a.x = s_val[t * 8 + 2 * k];
            a.y = s_val[t * 8 + 2 * k + 1];
            // (neg_a, A, neg_b, B, c_mod, C, reuse_a, reuse_b)
            c = __builtin_amdgcn_wmma_f32_16x16x4_f32(false, a, false, ones,
                                                      (short)0, c, false, false);
        }
        float s = c[0] + c[1] + c[2] + c[3] + c[4] + c[5] + c[6] + c[7];
        float s_hi = __shfl(s, 16, 32);
        if (t == 0) out[0] = mu * (s + s_hi);
#else
        float s = 0.0f;
#pragma unroll
        for (int k = 0; k < 8; ++k) s += s_val[t * 8 + k];
#pragma unroll
        for (int off = 16; off > 0; off >>= 1) s += __shfl_down(s, off, 32);
        if (t == 0) out[0] = mu * s;
#endif
    }
}

extern "C" void kernel_launch(void* const* d_in, const int* in_sizes, int n_in,
                              void* d_out, int out_size, void* d_ws, size_t ws_size,
                              hipStream_t stream) {
    const float*         pos          = (const float*)d_in[0];
    const int*           flat_netpin  = (const int*)d_in[1];
    const int*           netpin_start = (const int*)d_in[2];
    const unsigned char* net_mask     = (const unsigned char*)d_in[3];
    const int*           max_degree   = (const int*)d_in[4];

    const int num_nets = in_sizes[2] - 1;     // netpin_start has N+1 entries
    const int num_pins = in_sizes[0] / 2;     // pos is [2 * num_pins]
    const int nblocks  = (num_nets + NC_THREADS - 1) / NC_THREADS;

    float* partials = (float*)d_ws;           // nblocks floats of scratch

    nc_main_kernel<<<nblocks, NC_THREADS, 0, stream>>>(
        pos, flat_netpin, netpin_start, net_mask, max_degree,
        num_nets, num_pins, partials);
    nc_final_kernel<<<1, NC_THREADS, 0, stream>>>(
        partials, nblocks, (float*)d_out, NC_MU);
}